// ExtLinearAttention_73426760892657
// MI455X (gfx1250) — compile-verified
//
#include <hip/hip_runtime.h>

// ---------------- shapes ----------------
#define BB     8
#define NN     1024
#define CDIM   256
#define HEADS  4
#define DH     32
#define HID    128      // HEADS*DH
#define INDC   5
#define QSCALE 0.17677669529663688f   // 32^-0.5
#define LNEPS  1e-5f

typedef __attribute__((ext_vector_type(2))) float v2f;
typedef __attribute__((ext_vector_type(8))) float v8f;
typedef int async_v4i __attribute__((vector_size(16)));   // matches builtin param type

// Async global->LDS path (gfx1250, ASYNCcnt). Guarded: fall back to
// synchronous register staging if the builtin is unavailable.
#if defined(__has_builtin)
#  if __has_builtin(__builtin_amdgcn_global_load_async_to_lds_b128)
#    define USE_ASYNC_LDS 1
#  endif
#endif
#ifndef USE_ASYNC_LDS
#  define USE_ASYNC_LDS 0
#endif

__device__ __forceinline__ void wait_async_all() {
#if USE_ASYNC_LDS
#  if __has_builtin(__builtin_amdgcn_s_wait_asynccnt)
  __builtin_amdgcn_s_wait_asynccnt(0);
#  else
  asm volatile("s_wait_asynccnt 0x0" ::: "memory");
#  endif
#endif
}

// =====================================================================
// Kernel A: qkv = W_qkv(384x256) @ x(b,256,n)  ->  q,k,v (b,128,n) raw
// grid: 8*32 blocks of 256 threads; each block does one b and 32 columns
// =====================================================================
__global__ void __launch_bounds__(256) qkv_proj_kernel(
    const float* __restrict__ x, const float* __restrict__ Wqkv,
    float* __restrict__ qb, float* __restrict__ kb, float* __restrict__ vb) {
  __shared__ __align__(16) float xs[CDIM * 32];   // 32 KB
  const int b  = blockIdx.x >> 5;
  const int n0 = (blockIdx.x & 31) << 5;
  const int t  = threadIdx.x;
  // stage x[b, :, n0:n0+32] (8192 floats) via float4
  const float4* xg = reinterpret_cast<const float4*>(x + (size_t)b * CDIM * NN);
  for (int e = t; e < 2048; e += 256) {
    const int c = e >> 3, j4 = e & 7;
    reinterpret_cast<float4*>(xs)[c * 8 + j4] = xg[c * (NN / 4) + (n0 >> 2) + j4];
  }
  __syncthreads();
  const int j = t & 31, og = t >> 5;          // 8 output groups of 48 rows
  for (int r = 0; r < 48; ++r) {
    const int o = og * 48 + r;                // 0..383
    const float* wrow = Wqkv + (size_t)o * CDIM;
    float acc = 0.f;
#pragma unroll 8
    for (int c = 0; c < CDIM; ++c) acc = fmaf(wrow[c], xs[c * 32 + j], acc);
    float* dst = (o < HID) ? qb : ((o < 2 * HID) ? kb : vb);
    dst[((size_t)b * HID + (o % HID)) * NN + n0 + j] = acc;
  }
}

// =====================================================================
// Kernel B: softmax of q over d (32 values, stride NN), in place, * SCALE
// one thread per (b,h,n): 32768 threads
// =====================================================================
__global__ void __launch_bounds__(256) q_softmax_kernel(float* __restrict__ qb) {
  const int idx = blockIdx.x * 256 + threadIdx.x;     // b*h*n
  const int n = idx & (NN - 1);
  const int bh = idx >> 10;
  float* base = qb + (size_t)bh * DH * NN + n;
  float vals[DH];
  float m = -1e30f;
#pragma unroll
  for (int d = 0; d < DH; ++d) { vals[d] = base[(size_t)d * NN]; m = fmaxf(m, vals[d]); }
  float s = 0.f;
#pragma unroll
  for (int d = 0; d < DH; ++d) { vals[d] = __expf(vals[d] - m); s += vals[d]; }
  const float inv = QSCALE / s;
#pragma unroll
  for (int d = 0; d < DH; ++d) base[(size_t)d * NN] = vals[d] * inv;
}

// =====================================================================
// Kernel C: softmax of k over n (1024), in place. One block per (b,h,d) row.
// =====================================================================
__global__ void __launch_bounds__(256) k_softmax_kernel(float* __restrict__ kb) {
  __shared__ float red[256];
  float* base = kb + (size_t)blockIdx.x * NN;
  const int t = threadIdx.x;
  float v0 = base[t], v1 = base[t + 256], v2 = base[t + 512], v3 = base[t + 768];
  float m = fmaxf(fmaxf(v0, v1), fmaxf(v2, v3));
  red[t] = m; __syncthreads();
  for (int s = 128; s > 0; s >>= 1) { if (t < s) red[t] = fmaxf(red[t], red[t + s]); __syncthreads(); }
  m = red[0]; __syncthreads();
  v0 = __expf(v0 - m); v1 = __expf(v1 - m); v2 = __expf(v2 - m); v3 = __expf(v3 - m);
  red[t] = v0 + v1 + v2 + v3; __syncthreads();
  for (int s = 128; s > 0; s >>= 1) { if (t < s) red[t] += red[t + s]; __syncthreads(); }
  const float inv = 1.f / red[0];
  base[t] = v0 * inv; base[t + 256] = v1 * inv; base[t + 512] = v2 * inv; base[t + 768] = v3 * inv;
}

// =====================================================================
// Kernel D: kv[b,h,d,d'] = sum_j ks[b,h,d,j]*v[b,h,d',j]  (32x32 per head)
// one block per (b,h): 32 blocks of 256 threads, LDS-chunked over j
// =====================================================================
__global__ void __launch_bounds__(256) kv_kernel(
    const float* __restrict__ ks, const float* __restrict__ vb, float* __restrict__ kv) {
  __shared__ __align__(16) float sk[DH * 128];
  __shared__ __align__(16) float sv[DH * 128];
  const int bh = blockIdx.x, t = threadIdx.x;
  const float* kbase = ks + (size_t)bh * DH * NN;
  const float* vbase = vb + (size_t)bh * DH * NN;
  const int d = t >> 3, dp0 = (t & 7) * 4;
  float acc[4] = {0.f, 0.f, 0.f, 0.f};
  for (int j0 = 0; j0 < NN; j0 += 128) {
    __syncthreads();
    for (int e = t; e < 1024; e += 256) {     // 1024 float4 per buffer
      const int r = e >> 5, c4 = e & 31;
      reinterpret_cast<float4*>(sk)[r * 32 + c4] =
          reinterpret_cast<const float4*>(kbase + (size_t)r * NN + j0)[c4];
      reinterpret_cast<float4*>(sv)[r * 32 + c4] =
          reinterpret_cast<const float4*>(vbase + (size_t)r * NN + j0)[c4];
    }
    __syncthreads();
#pragma unroll 4
    for (int j = 0; j < 128; ++j) {
      const float kval = sk[d * 128 + j];
      acc[0] = fmaf(kval, sv[(dp0 + 0) * 128 + j], acc[0]);
      acc[1] = fmaf(kval, sv[(dp0 + 1) * 128 + j], acc[1]);
      acc[2] = fmaf(kval, sv[(dp0 + 2) * 128 + j], acc[2]);
      acc[3] = fmaf(kval, sv[(dp0 + 3) * 128 + j], acc[3]);
    }
  }
  float* out = kv + (size_t)bh * DH * DH + d * DH + dp0;
  out[0] = acc[0]; out[1] = acc[1]; out[2] = acc[2]; out[3] = acc[3];
}

// =====================================================================
// Kernel E (WMMA + async-to-LDS): out_pre_T[b][i][h*32+d'] =
//   sum_j qkbias[b,h,i,j]*v[b,h,d',j]  +  sum_d qs[b,h,d,i]*kv[b,h,d,d']
// qkbias computed on the fly from indicator tile; indicator is streamed
// with double-buffered GLOBAL_LOAD_ASYNC_TO_LDS_B128 (ASYNCcnt) so the
// HBM stream of chunk j+1 overlaps the WMMA work on chunk j.
// grid: 8*64 blocks, 128 threads (4 waves = 4 heads). i-tile = 16 rows.
// =====================================================================
__global__ void __launch_bounds__(128) attn_wmma_kernel(
    const float* __restrict__ ind, const float* __restrict__ Wind,
    const float* __restrict__ qs, const float* __restrict__ vb,
    const float* __restrict__ kv, float* __restrict__ outp) {
  __shared__ __align__(16) float sind[2][INDC * 16 * 64];   // 2 x 20 KB
  const int b  = blockIdx.x >> 6;
  const int i0 = (blockIdx.x & 63) << 4;
  const int t  = threadIdx.x;
  const int h = t >> 5, lane = t & 31;
  const int row = lane & 15;            // A-matrix M row / B-matrix N col
  const int kb2 = (lane >> 4) << 1;     // K lane-group offset: 0 or 2
  const float w0 = Wind[h * INDC + 0], w1 = Wind[h * INDC + 1],
              w2 = Wind[h * INDC + 2], w3 = Wind[h * INDC + 3],
              w4 = Wind[h * INDC + 4];
  const float* vh   = vb + ((size_t)b * HEADS + h) * DH * NN;
  const float* qh   = qs + ((size_t)b * HEADS + h) * DH * NN;
  const float* kvh  = kv + ((size_t)b * HEADS + h) * DH * DH;
  const float* indb = ind + (size_t)b * INDC * NN * NN;

  // stage indicator[b, 0..4, i0:i0+16, jc:jc+64] into LDS buffer `buf`
  auto stage = [&](int buf, int jc) {
    for (int e = t; e < 1280; e += 128) {       // 1280 x float4
      const int c = e >> 8, rem = e & 255, r = rem >> 4, j4 = rem & 15;
      const float* gp = indb + ((size_t)c * NN + i0 + r) * NN + jc + j4 * 4;
      float* lp = &sind[buf][e * 4];
#if USE_ASYNC_LDS
      __builtin_amdgcn_global_load_async_to_lds_b128(
          (__attribute__((address_space(1))) async_v4i*)const_cast<float*>(gp),
          (__attribute__((address_space(3))) async_v4i*)lp, 0, 0);
#else
      *reinterpret_cast<float4*>(lp) = *reinterpret_cast<const float4*>(gp);
#endif
    }
  };

  v8f acc0 = {};   // d' tile 0..15
  v8f acc1 = {};   // d' tile 16..31

  stage(0, 0);                                  // prologue: chunk 0
  for (int jc = 0; jc < NN; jc += 64) {
    const int cur = (jc >> 6) & 1;
    wait_async_all();                           // my loads into buf[cur] landed
    __syncthreads();                            // everyone's landed + buf[cur^1] free
    if (jc + 64 < NN) stage(cur ^ 1, jc + 64);  // overlap next chunk's DMA
    const float2* sp = reinterpret_cast<const float2*>(&sind[cur][0]);
#pragma unroll
    for (int kk = 0; kk < 16; ++kk) {
      const int j0 = kk * 4 + kb2;
      const int base2 = row * 32 + (j0 >> 1);   // float2 index in c-plane (512/plane)
      const float2 p0 = sp[base2];
      const float2 p1 = sp[base2 + 512];
      const float2 p2 = sp[base2 + 1024];
      const float2 p3 = sp[base2 + 1536];
      const float2 p4 = sp[base2 + 2048];
      v2f a;
      a.x = fmaf(w4, p4.x, fmaf(w3, p3.x, fmaf(w2, p2.x, fmaf(w1, p1.x, w0 * p0.x))));
      a.y = fmaf(w4, p4.y, fmaf(w3, p3.y, fmaf(w2, p2.y, fmaf(w1, p1.y, w0 * p0.y))));
      const float2 bv0 = *reinterpret_cast<const float2*>(vh + (size_t)row * NN + jc + j0);
      const float2 bv1 = *reinterpret_cast<const float2*>(vh + (size_t)(row + 16) * NN + jc + j0);
      v2f bf0; bf0.x = bv0.x; bf0.y = bv0.y;
      v2f bf1; bf1.x = bv1.x; bf1.y = bv1.y;
      acc0 = __builtin_amdgcn_wmma_f32_16x16x4_f32(false, a, false, bf0, (short)0, acc0, false, false);
      acc1 = __builtin_amdgcn_wmma_f32_16x16x4_f32(false, a, false, bf1, (short)0, acc1, false, false);
    }
  }

  // out1: A[i,d] = qs[d,i0+row], B[d,d'] = kv[d,d'];  K = 32
#pragma unroll
  for (int k0 = 0; k0 < DH; k0 += 4) {
    const int kd = k0 + kb2;
    v2f a, bf0, bf1;
    a.x = qh[(size_t)kd * NN + i0 + row];
    a.y = qh[(size_t)(kd + 1) * NN + i0 + row];
    bf0.x = kvh[kd * DH + row];       bf0.y = kvh[(kd + 1) * DH + row];
    bf1.x = kvh[kd * DH + row + 16];  bf1.y = kvh[(kd + 1) * DH + row + 16];
    acc0 = __builtin_amdgcn_wmma_f32_16x16x4_f32(false, a, false, bf0, (short)0, acc0, false, false);
    acc1 = __builtin_amdgcn_wmma_f32_16x16x4_f32(false, a, false, bf1, (short)0, acc1, false, false);
  }

  // store transposed: out_pre_T[b][i][h*32+d']  (coalesced over d')
#pragma unroll
  for (int r = 0; r < 8; ++r) {
    const int i = i0 + r + ((lane >> 4) << 3);
    float* orow = outp + ((size_t)b * NN + i) * HID + h * DH;
    orow[row]      = acc0[r];
    orow[row + 16] = acc1[r];
  }
}

// =====================================================================
// Kernel F: y = W_out(256x128) @ out_pre + b_out, then LayerNorm over 256 ch.
// grid 8*32 blocks, 256 threads; 32 columns per block, coalesced output.
// =====================================================================
__global__ void __launch_bounds__(256) out_ln_kernel(
    const float* __restrict__ outp, const float* __restrict__ Wout,
    const float* __restrict__ bout, const float* __restrict__ g,
    float* __restrict__ y) {
  __shared__ __align__(16) float scol[HID * 32];   // 16 KB [c][nl]
  __shared__ float sy[CDIM * 32];                  // 32 KB [o][nl]
  __shared__ float sredA[8 * 32];
  __shared__ float sredB[8 * 32];
  const int b  = blockIdx.x >> 5;
  const int n0 = (blockIdx.x & 31) << 5;
  const int t  = threadIdx.x;
  const int nl = t & 31, og = t >> 5;
  // stage out_pre_T[b, n0+nl, :] : each thread loads 16 channels (4 float4)
  const float4* src = reinterpret_cast<const float4*>(outp + ((size_t)b * NN + n0 + nl) * HID);
#pragma unroll
  for (int q = 0; q < 4; ++q) {
    const float4 vv = src[og * 4 + q];
    const int c = og * 16 + q * 4;
    scol[(c + 0) * 32 + nl] = vv.x; scol[(c + 1) * 32 + nl] = vv.y;
    scol[(c + 2) * 32 + nl] = vv.z; scol[(c + 3) * 32 + nl] = vv.w;
  }
  __syncthreads();
  float part = 0.f, part2 = 0.f;
  for (int oo = 0; oo < 32; ++oo) {
    const int o = og * 32 + oo;
    const float* wrow = Wout + (size_t)o * HID;
    float acc = bout[o];
#pragma unroll 8
    for (int c = 0; c < HID; ++c) acc = fmaf(wrow[c], scol[c * 32 + nl], acc);
    sy[o * 32 + nl] = acc;
    part += acc; part2 += acc * acc;
  }
  sredA[og * 32 + nl] = part;
  sredB[og * 32 + nl] = part2;
  __syncthreads();
  if (t < 32) {
    float s = 0.f, s2 = 0.f;
    for (int r = 0; r < 8; ++r) { s += sredA[r * 32 + t]; s2 += sredB[r * 32 + t]; }
    const float mean = s * (1.f / CDIM);
    const float var  = s2 * (1.f / CDIM) - mean * mean;
    sredA[t] = mean;
    sredB[t] = rsqrtf(var + LNEPS);
  }
  __syncthreads();
  const float mean = sredA[nl], inv = sredB[nl];
  for (int oo = 0; oo < 32; ++oo) {
    const int o = og * 32 + oo;
    y[((size_t)b * CDIM + o) * NN + n0 + nl] = (sy[o * 32 + nl] - mean) * inv * g[o];
  }
}

// =====================================================================
extern "C" void kernel_launch(void* const* d_in, const int* in_sizes, int n_in,
                              void* d_out, int out_size, void* d_ws, size_t ws_size,
                              hipStream_t stream) {
  const float* x     = (const float*)d_in[0];
  const float* ind   = (const float*)d_in[1];
  const float* Wqkv  = (const float*)d_in[2];
  const float* Wind  = (const float*)d_in[3];
  const float* Wout  = (const float*)d_in[4];
  const float* bout  = (const float*)d_in[5];
  const float* g     = (const float*)d_in[6];
  float* out = (float*)d_out;

  const size_t Q = (size_t)BB * HID * NN;        // 1,048,576 floats
  float* ws   = (float*)d_ws;
  float* qb   = ws;                  // q raw -> softmaxed*scale in place
  float* kb   = ws + Q;              // k raw -> softmaxed in place
  float* vb   = ws + 2 * Q;          // v
  float* kvb  = ws + 3 * Q;          // (b,h,32,32) = 32768 floats
  float* outp = ws + 3 * Q + 32768;  // out_pre_T (b, n, 128)

  qkv_proj_kernel<<<BB * 32, 256, 0, stream>>>(x, Wqkv, qb, kb, vb);
  q_softmax_kernel<<<(BB * HEADS * NN) / 256, 256, 0, stream>>>(qb);
  k_softmax_kernel<<<BB * HEADS * DH, 256, 0, stream>>>(kb);
  kv_kernel<<<BB * HEADS, 256, 0, stream>>>(kb, vb, kvb);
  attn_wmma_kernel<<<BB * 64, 128, 0, stream>>>(ind, Wind, qb, vb, kvb, outp);
  out_ln_kernel<<<BB * 32, 256, 0, stream>>>(outp, Wout, bout, g, out);
}